// RotatE_71253507441336
// MI455X (gfx1250) — compile-verified
//
#include <hip/hip_runtime.h>
#include <math.h>

#define NUM_ENTITIES 14541
#define EMBED_DIM    500
#define NB           8           // batch size B
#define GAMMA_F      12.0f
// EMBEDDING_RANGE = (12+2)/500 = 0.028 ; phase scale = PI / 0.028
#define PHASE_SCALE  112.19973762821175f

typedef __attribute__((ext_vector_type(4))) unsigned int v4u;
typedef __attribute__((ext_vector_type(8))) int          v8i;
typedef __attribute__((ext_vector_type(4))) int          v4i;

// ---------------------------------------------------------------------------
// Kernel 1: precompute A[b,d] = re_h*cos - im_h*sin, C[b,d] = re_h*sin + im_h*cos
// into workspace: AC[0..3999] = A, AC[4000..7999] = C  (32 KB total)
// ---------------------------------------------------------------------------
__global__ __launch_bounds__(256) void rotate_ac_kernel(
    const int* __restrict__ h_idx, const int* __restrict__ r_idx,
    const float* __restrict__ E, const float* __restrict__ R,
    float* __restrict__ AC)
{
    int i = blockIdx.x * blockDim.x + threadIdx.x;
    if (i >= NB * EMBED_DIM) return;
    int b = i / EMBED_DIM;
    int d = i - b * EMBED_DIM;
    int h = h_idx[b];
    int r = r_idx[b];
    float rh = E[(size_t)h * (2 * EMBED_DIM) + d];
    float ih = E[(size_t)h * (2 * EMBED_DIM) + EMBED_DIM + d];
    float phase = R[(size_t)r * EMBED_DIM + d] * PHASE_SCALE;
    float cr = cosf(phase);
    float sr = sinf(phase);
    AC[i]                    = rh * cr - ih * sr;   // A
    AC[NB * EMBED_DIM + i]   = rh * sr + ih * cr;   // C
}

// ---------------------------------------------------------------------------
// Kernel 2: score all entities. Block = 256 threads (8 waves), each wave
// handles 4 entities, lanes striped over d in float4 chunks (125 chunks).
// A/C table (32 KB) staged into LDS via TDM tensor_load_to_lds (+ safe
// cooperative overwrite), then streamed compute with 8 accumulators/lane.
// ---------------------------------------------------------------------------
#define EPW 4                    // entities per wave
#define EPB (8 * EPW)            // entities per block = 32
#define NCHUNK (EMBED_DIM / 4)   // 125 float4 chunks per half-row

__device__ __forceinline__ float dist4(float4 a, float4 c, float4 rt, float4 it)
{
    float rx = a.x - rt.x, ix = c.x - it.x;
    float ry = a.y - rt.y, iy = c.y - it.y;
    float rz = a.z - rt.z, iz = c.z - it.z;
    float rw = a.w - rt.w, iw = c.w - it.w;
    float s0 = __builtin_amdgcn_sqrtf(__builtin_fmaf(rx, rx, ix * ix));
    float s1 = __builtin_amdgcn_sqrtf(__builtin_fmaf(ry, ry, iy * iy));
    float s2 = __builtin_amdgcn_sqrtf(__builtin_fmaf(rz, rz, iz * iz));
    float s3 = __builtin_amdgcn_sqrtf(__builtin_fmaf(rw, rw, iw * iw));
    return (s0 + s1) + (s2 + s3);
}

__global__ __launch_bounds__(256) void rotate_score_kernel(
    const float* __restrict__ E, const float* __restrict__ AC,
    const float* __restrict__ bias, float* __restrict__ out)
{
    __shared__ __align__(16) float sAC[2 * NB * EMBED_DIM];   // 8000 floats = 32 KB

    // --- Stage AC table into LDS with the Tensor Data Mover (wave 0 only).
    // D# per CDNA5 ISA ch.8: 1 descriptor, type=2, data_size=4B,
    // 1-row tile of 8000 elements (tile_dim0=8000 <= 64K).
    if (threadIdx.x < 32) {
        unsigned long long g = (unsigned long long)(uintptr_t)AC;
        unsigned int lds = (unsigned int)(uintptr_t)(&sAC[0]);   // low 32 bits = LDS byte offset
        v4u g0;
        g0[0] = 1u;                                   // count=1 valid, user mode
        g0[1] = lds;                                  // lds_addr
        g0[2] = (unsigned int)g;                      // global_addr[31:0]
        g0[3] = (unsigned int)((g >> 32) & 0x1FFFFFFu) | 0x80000000u; // addr[56:32] | type=2
        v8i g1;
        g1[0] = 0x00020000;       // data_size=2 (4 bytes), no multicast/pad/iterate
        g1[1] = 0x1F400000;       // tensor_dim0[15:0]=8000 in bits[63:48]
        g1[2] = 0x00010000;       // tensor_dim0[31:16]=0, tensor_dim1=1
        g1[3] = 0x1F400000;       // tile_dim0=8000 in bits[127:112]
        g1[4] = 0x00000001;       // tile_dim1=1, tile_dim2=0
        g1[5] = 8000;             // tensor_dim0_stride[31:0]
        g1[6] = 0x1F400000;       // dim0_stride[47:32]=0, tensor_dim1_stride[15:0]=8000
        g1[7] = 0;                // tensor_dim1_stride[47:16]=0
        v4i z4; z4[0] = 0; z4[1] = 0; z4[2] = 0; z4[3] = 0;
        v8i z8; z8[0] = 0; z8[1] = 0; z8[2] = 0; z8[3] = 0;
        z8[4] = 0; z8[5] = 0; z8[6] = 0; z8[7] = 0;
        // 6-arg form (clang-23 / therock-10.0 headers toolchain)
        __builtin_amdgcn_tensor_load_to_lds(g0, g1, z4, z4, z8, 0);
        __builtin_amdgcn_s_wait_tensorcnt(0);
    }
    __syncthreads();
    // Safety net: cooperative overwrite with identical data (TDM semantics are
    // probe-unverified; OOB LDS writes drop, so final LDS state is correct
    // whether or not the TDM transfer landed).
    for (int i = threadIdx.x; i < 2 * NB * EMBED_DIM; i += 256)
        sAC[i] = AC[i];
    __syncthreads();

    const float* sA = sAC;
    const float* sC = sAC + NB * EMBED_DIM;

    const int wave = threadIdx.x >> 5;
    const int lane = threadIdx.x & 31;
    const int e0 = blockIdx.x * EPB + wave * EPW;

    for (int i = 0; i < EPW; ++i) {
        const int e = e0 + i;               // wave-uniform
        if (e >= NUM_ENTITIES) break;

        const float4* reRow = (const float4*)(E + (size_t)e * (2 * EMBED_DIM));
        const float4* imRow = (const float4*)(E + (size_t)e * (2 * EMBED_DIM) + EMBED_DIM);

        float acc[NB];
#pragma unroll
        for (int b = 0; b < NB; ++b) acc[b] = 0.0f;

        for (int c = lane; c < NCHUNK; c += 32) {
            float4 rt = reRow[c];           // coalesced global_load_b128
            float4 it = imRow[c];
            const int d4 = c * 4;
#pragma unroll
            for (int b = 0; b < NB; ++b) {
                float4 a  = *(const float4*)(sA + b * EMBED_DIM + d4);  // ds_load_b128
                float4 cc = *(const float4*)(sC + b * EMBED_DIM + d4);
                acc[b] += dist4(a, cc, rt, it);
            }
        }

        // wave32 butterfly reduction over lanes for all 8 accumulators
#pragma unroll
        for (int off = 16; off > 0; off >>= 1) {
#pragma unroll
            for (int b = 0; b < NB; ++b)
                acc[b] += __shfl_xor(acc[b], off, 32);
        }

        if (lane == 0) {
            float bb = bias[e];
#pragma unroll
            for (int b = 0; b < NB; ++b) {
                float s = GAMMA_F - acc[b] + bb;
                out[(size_t)b * NUM_ENTITIES + e] = 1.0f / (1.0f + expf(-s));
            }
        }
    }
}

// ---------------------------------------------------------------------------
extern "C" void kernel_launch(void* const* d_in, const int* in_sizes, int n_in,
                              void* d_out, int out_size, void* d_ws, size_t ws_size,
                              hipStream_t stream)
{
    const int*   h_idx = (const int*)d_in[0];
    const int*   r_idx = (const int*)d_in[1];
    const float* E     = (const float*)d_in[2];
    const float* R     = (const float*)d_in[3];
    const float* bias  = (const float*)d_in[4];
    float*       out   = (float*)d_out;
    float*       AC    = (float*)d_ws;     // needs 8000 floats = 32 KB

    (void)in_sizes; (void)n_in; (void)out_size; (void)ws_size;

    const int ac_elems  = NB * EMBED_DIM;                       // 4000
    const int ac_blocks = (ac_elems + 255) / 256;               // 16
    rotate_ac_kernel<<<ac_blocks, 256, 0, stream>>>(h_idx, r_idx, E, R, AC);

    const int score_blocks = (NUM_ENTITIES + EPB - 1) / EPB;    // 455
    rotate_score_kernel<<<score_blocks, 256, 0, stream>>>(E, AC, bias, out);
}